// H_1864015806665
// MI455X (gfx1250) — compile-verified
//
#include <hip/hip_runtime.h>

typedef float v2f __attribute__((ext_vector_type(2)));
typedef float v8f __attribute__((ext_vector_type(8)));

#define N_TOT      4096      // 4^L
#define RDIM       32
#define BDIM       16        // batch rows -> WMMA M
#define S          128       // 4*r -> K and N of per-block matmul
#define M_BLOCKS   1024      // N_TOT / 4
#define LDS_STRIDE 132       // 128 + 4 pad (floats) to spread LDS banks

// One block per 'a'. 512 threads = 16 waves:
//   wave = (c<<3) | nt ; c: 0 -> Y_re, 1 -> Y_im ; nt: 16-col output tile.
__launch_bounds__(512, 1)
__global__ void hmat_wmma_f32_kernel(const float* __restrict__ x,
                                     const float* __restrict__ hr1,
                                     const float* __restrict__ hi1,
                                     const float* __restrict__ hr2,
                                     const float* __restrict__ hi2,
                                     const int*   __restrict__ perm,
                                     float* __restrict__ out)
{
    __shared__ float ldsX[2 * BDIM * LDS_STRIDE]; // X_re / X_im, padded rows
    __shared__ int   ldsPerm[4];

    const int a   = blockIdx.x;
    const int tid = threadIdx.x;

    if (tid < 4) ldsPerm[tid] = perm[a * 4 + tid];
    __syncthreads();

    // ---- Gather permuted x rows into LDS (2*16*128 floats, float4 coalesced) ----
    // xp[b, c, a, j] = x[b, c, perm[a*4 + j/32], j%32]
    #pragma unroll
    for (int i = 0; i < 2; ++i) {
        int e4 = i * 512 + tid;            // float4 granule id: 0..1023
        int c  = e4 >> 9;                  // 0..1
        int b  = (e4 >> 5) & 15;           // 0..15
        int j4 = e4 & 31;                  // 32 float4 per row (j = j4*4)
        int q  = j4 >> 3;                  // which of the 4 permuted source rows
        int rr = (j4 & 7) * 4;             // offset within r=32
        int n  = ldsPerm[q];
        const float4 v = *reinterpret_cast<const float4*>(
            x + ((size_t)(b * 2 + c) * N_TOT + (size_t)n) * RDIM + rr);
        *reinterpret_cast<float4*>(
            &ldsX[c * (BDIM * LDS_STRIDE) + b * LDS_STRIDE + j4 * 4]) = v;
    }
    __syncthreads();

    // ---- Per-wave 16x16 output tile via V_WMMA_F32_16X16X4_F32 ----
    const int wave = tid >> 5;
    const int lane = tid & 31;
    const int c    = wave >> 3;        // 0: real output, 1: imag output
    const int nt   = wave & 7;         // output column tile
    const int half = lane >> 4;        // lane group: selects K pair / M high half
    const int ml   = lane & 15;        // M for A-frag, N-in-tile for B/D frags

    // Y_re = X_re*Hr1 + X_im*Hi1 ; Y_im = X_re*Hi2 + X_im*Hr2
    const float* __restrict__ W0 = (c == 0 ? hr1 : hi2) + (size_t)a * S * S;
    const float* __restrict__ W1 = (c == 0 ? hi1 : hr2) + (size_t)a * S * S;

    const int col = nt * 16 + ml;      // global output column for this lane
    const float* are_base = &ldsX[0 * (BDIM * LDS_STRIDE) + ml * LDS_STRIDE + 2 * half];
    const float* aim_base = &ldsX[1 * (BDIM * LDS_STRIDE) + ml * LDS_STRIDE + 2 * half];

    v8f acc = {};
    #pragma unroll 8
    for (int k = 0; k < S; k += 4) {
        // A fragments (16x4): lane holds X[ml][k+2*half], X[ml][k+2*half+1]
        v2f ar = *reinterpret_cast<const v2f*>(are_base + k);
        v2f ai = *reinterpret_cast<const v2f*>(aim_base + k);

        // B fragments (4x16): VGPR0 = rows k+2*half, VGPR1 = rows k+2*half+1,
        // cols nt*16..nt*16+15 across lanes 0-15 / 16-31. Streamed NT (no reuse).
        const int row = k + 2 * half;
        v2f b0, b1;
        b0.x = __builtin_nontemporal_load(W0 + (size_t)row * S + col);
        b0.y = __builtin_nontemporal_load(W0 + (size_t)(row + 1) * S + col);
        b1.x = __builtin_nontemporal_load(W1 + (size_t)row * S + col);
        b1.y = __builtin_nontemporal_load(W1 + (size_t)(row + 1) * S + col);

        acc = __builtin_amdgcn_wmma_f32_16x16x4_f32(false, ar, false, b0,
                                                    (short)0, acc, false, false);
        acc = __builtin_amdgcn_wmma_f32_16x16x4_f32(false, ai, false, b1,
                                                    (short)0, acc, false, false);
    }

    // ---- Store D tile: VGPR v -> M = v + 8*half, N = col ----
    // out[b, c, a, j] flat = (b*2+c)*131072 + a*128 + j (trailing reshape is linear)
    #pragma unroll
    for (int v = 0; v < 8; ++v) {
        const int M = v + half * 8;
        out[(size_t)(M * 2 + c) * (N_TOT * RDIM) + (size_t)a * S + col] = acc[v];
    }
}

extern "C" void kernel_launch(void* const* d_in, const int* in_sizes, int n_in,
                              void* d_out, int out_size, void* d_ws, size_t ws_size,
                              hipStream_t stream) {
    const float* x    = (const float*)d_in[0];
    const float* hr1  = (const float*)d_in[1];
    const float* hi1  = (const float*)d_in[2];
    const float* hr2  = (const float*)d_in[3];
    const float* hi2  = (const float*)d_in[4];
    const int*   perm = (const int*)d_in[5];
    float* out = (float*)d_out;

    hmat_wmma_f32_kernel<<<M_BLOCKS, 512, 0, stream>>>(x, hr1, hi1, hr2, hi2, perm, out);
}